// SVDHead_64330020159934
// MI455X (gfx1250) — compile-verified
//
#include <hip/hip_runtime.h>
#include <math.h>

// ---------------------------------------------------------------------------
// SVDHead pipeline for MI455X (gfx1250, wave32).
// Heavy op: C[b,n,m] = sum_d srcE[b,d,n]*tgtE[b,d,m]  (8.6 GFLOP f32)
//  -> V_WMMA_F32_16X16X4_F32, 2x2 C-tiles per wave (1.0 LDS-load/WMMA).
// The second big GEMM (src_corr_emb) is eliminated algebraically:
//  diag[b,n] = tau[b] * sum_j perm_norm[b,n,j] * affinity0[b,n,j].
// Gumbel straight-through == hard one-hot in forward -> argmax + gather.
// ---------------------------------------------------------------------------

typedef __attribute__((ext_vector_type(2))) float v2f;
typedef __attribute__((ext_vector_type(8))) float v8f;

#define B_   8
#define D_   512
#define N_   1024
#define K_   256
#define NP   1025
#define BN_EPS 1e-5f
#define RSQRT_D 0.044194173824159216f   // 1/sqrt(512)

// -------------------- zero the Sinkhorn pad row/col ------------------------
__global__ void pad_zero_kernel(float* __restrict__ spad) {
  const int b = blockIdx.y;
  const int i = blockIdx.x * blockDim.x + threadIdx.x;
  if (i < NP) {
    float* s = spad + (size_t)b * NP * NP;
    s[(size_t)N_ * NP + i] = 0.0f;   // pad row
    s[(size_t)i * NP + N_] = 0.0f;   // pad col
  }
}

// -------------------- affinity GEMM (WMMA f32 16x16x4) ---------------------
// block tile: 64 rows (n) x 128 cols (m); 8 waves in 2x4 grid, each wave a
// 32x32 super-tile = 2x2 WMMA tiles -> 4 fragment loads per 4 WMMAs.
__global__ __launch_bounds__(256)
void affinity_gemm_kernel(const float* __restrict__ srcE,
                          const float* __restrict__ tgtE,
                          const float* __restrict__ temperature,
                          float* __restrict__ aff0,   // [B][N][N]
                          float* __restrict__ spad)   // [B][NP][NP]
{
  __shared__ __align__(16) float As[64][18];    // [n][k], stride 18 (even)
  __shared__ __align__(16) float Bs[128][18];   // [m][k]

  const int b  = blockIdx.z;
  const int n0 = blockIdx.y * 64;
  const int m0 = blockIdx.x * 128;
  const int t  = threadIdx.x;
  const int lane  = t & 31;
  const int wave  = t >> 5;            // 0..7
  const int wr    = wave & 1;          // 32-row group 0..1
  const int wc    = wave >> 1;         // 32-col group 0..3
  const int l15   = lane & 15;
  const int khalf = (lane >> 4) * 2;   // ISA A/B frag: lanes>=16 hold K+2,K+3

  const float* Ab = srcE + (size_t)b * D_ * N_;
  const float* Bb = tgtE + (size_t)b * D_ * N_;

  const int lk = t >> 4;               // k-row this thread stages (0..15)
  const int ln = (t & 15) * 4;         // float4 column offset

  v8f c00 = {}; v8f c01 = {}; v8f c10 = {}; v8f c11 = {};

  for (int kb = 0; kb < D_; kb += 16) {
    __syncthreads();
    const float4 av = *(const float4*)(Ab + (size_t)(kb + lk) * N_ + n0 + ln);
    As[ln + 0][lk] = av.x; As[ln + 1][lk] = av.y;
    As[ln + 2][lk] = av.z; As[ln + 3][lk] = av.w;
    const float4 bv0 = *(const float4*)(Bb + (size_t)(kb + lk) * N_ + m0 + ln);
    Bs[ln + 0][lk] = bv0.x; Bs[ln + 1][lk] = bv0.y;
    Bs[ln + 2][lk] = bv0.z; Bs[ln + 3][lk] = bv0.w;
    const float4 bv1 = *(const float4*)(Bb + (size_t)(kb + lk) * N_ + m0 + 64 + ln);
    Bs[64 + ln + 0][lk] = bv1.x; Bs[64 + ln + 1][lk] = bv1.y;
    Bs[64 + ln + 2][lk] = bv1.z; Bs[64 + ln + 3][lk] = bv1.w;
    if (kb + 16 < D_) {   // pull next K-chunk toward L2 (global_prefetch_b8)
      __builtin_prefetch(Ab + (size_t)(kb + 16 + lk) * N_ + n0 + ln, 0, 1);
      __builtin_prefetch(Bb + (size_t)(kb + 16 + lk) * N_ + m0 + ln, 0, 1);
      __builtin_prefetch(Bb + (size_t)(kb + 16 + lk) * N_ + m0 + 64 + ln, 0, 1);
    }
    __syncthreads();
#pragma unroll
    for (int k0 = 0; k0 < 16; k0 += 4) {
      const v2f a0 = *(const v2f*)&As[wr * 32 + l15][k0 + khalf];
      const v2f a1 = *(const v2f*)&As[wr * 32 + 16 + l15][k0 + khalf];
      const v2f b0 = *(const v2f*)&Bs[wc * 32 + l15][k0 + khalf];
      const v2f b1 = *(const v2f*)&Bs[wc * 32 + 16 + l15][k0 + khalf];
      c00 = __builtin_amdgcn_wmma_f32_16x16x4_f32(false, a0, false, b0,
                                                  (short)0, c00, false, false);
      c01 = __builtin_amdgcn_wmma_f32_16x16x4_f32(false, a0, false, b1,
                                                  (short)0, c01, false, false);
      c10 = __builtin_amdgcn_wmma_f32_16x16x4_f32(false, a1, false, b0,
                                                  (short)0, c10, false, false);
      c11 = __builtin_amdgcn_wmma_f32_16x16x4_f32(false, a1, false, b1,
                                                  (short)0, c11, false, false);
    }
  }

  const float scale = RSQRT_D / temperature[b];   // (1/sqrt(D)) / tau
  const int rbase = n0 + wr * 32 + ((lane >> 4) * 8);
  const int cbase = m0 + wc * 32 + l15;
  float* aout = aff0 + (size_t)b * N_ * N_;
  float* sout = spad + (size_t)b * NP * NP;
#pragma unroll
  for (int v = 0; v < 8; ++v) {
    const int rA = rbase + v;
    const int rB = rA + 16;
    float x;
    x = c00[v] * scale;
    aout[(size_t)rA * N_ + cbase]      = x; sout[(size_t)rA * NP + cbase]      = x;
    x = c01[v] * scale;
    aout[(size_t)rA * N_ + cbase + 16] = x; sout[(size_t)rA * NP + cbase + 16] = x;
    x = c10[v] * scale;
    aout[(size_t)rB * N_ + cbase]      = x; sout[(size_t)rB * NP + cbase]      = x;
    x = c11[v] * scale;
    aout[(size_t)rB * N_ + cbase + 16] = x; sout[(size_t)rB * NP + cbase + 16] = x;
  }
}

// -------------------- Sinkhorn row normalize (rows 0..N-1, all NP cols) ----
__global__ __launch_bounds__(256)
void sinkhorn_row_kernel(float* __restrict__ spad) {
  __shared__ float sm[256], ss[256];
  const int b = blockIdx.y, i = blockIdx.x, t = threadIdx.x;
  float* row = spad + (size_t)b * NP * NP + (size_t)i * NP;
  float m = -3.4e38f, s = 0.0f;
  for (int j = t; j < NP; j += 256) {
    const float v = row[j];
    const float nm = fmaxf(m, v);
    s = s * __expf(m - nm) + __expf(v - nm);
    m = nm;
  }
  sm[t] = m; ss[t] = s;
  __syncthreads();
  for (int off = 128; off > 0; off >>= 1) {
    if (t < off) {
      const float m2 = sm[t + off], s2 = ss[t + off];
      const float nm = fmaxf(sm[t], m2);
      ss[t] = ss[t] * __expf(sm[t] - nm) + s2 * __expf(m2 - nm);
      sm[t] = nm;
    }
    __syncthreads();
  }
  const float lse = sm[0] + __logf(ss[0]);
  for (int j = t; j < NP; j += 256) row[j] -= lse;
}

// -------------------- Sinkhorn col normalize (cols 0..N-1, all NP rows) ----
// block = (32 cols) x (8 row-walkers); coalesced 128B row segments
__global__ __launch_bounds__(256)
void sinkhorn_col_kernel(float* __restrict__ spad) {
  __shared__ float sm[8][32], ss[8][32], lse[32];
  const int b  = blockIdx.y;
  const int j0 = blockIdx.x * 32;
  const int x  = threadIdx.x & 31;
  const int y  = threadIdx.x >> 5;
  float* s = spad + (size_t)b * NP * NP;
  const int j = j0 + x;
  float m = -3.4e38f, acc = 0.0f;
  for (int i = y; i < NP; i += 8) {
    const float v = s[(size_t)i * NP + j];
    const float nm = fmaxf(m, v);
    acc = acc * __expf(m - nm) + __expf(v - nm);
    m = nm;
  }
  sm[y][x] = m; ss[y][x] = acc;
  __syncthreads();
  if (y == 0) {
    float M = sm[0][x], S = ss[0][x];
#pragma unroll
    for (int yy = 1; yy < 8; ++yy) {
      const float m2 = sm[yy][x], s2 = ss[yy][x];
      const float nm = fmaxf(M, m2);
      S = S * __expf(M - nm) + s2 * __expf(m2 - nm);
      M = nm;
    }
    lse[x] = M + __logf(S);
  }
  __syncthreads();
  const float L = lse[x];
  for (int i = y; i < NP; i += 8) s[(size_t)i * NP + j] -= L;
}

// ------ fused: perm_norm + diag dot(P,aff0)*tau + src_corr = P @ tgt^T -----
__global__ __launch_bounds__(256)
void permnorm_kernel(const float* __restrict__ spad,
                     const float* __restrict__ aff0,
                     const float* __restrict__ tgt,         // [B][3][N]
                     const float* __restrict__ temperature,
                     float* __restrict__ perm_out,          // [B][N][N]
                     float* __restrict__ diag,              // [B][N]
                     float* __restrict__ src_corr)          // [B][3][N]
{
  __shared__ float red[4][256];
  const int b = blockIdx.y, i = blockIdx.x, t = threadIdx.x;
  const float* lrow = spad + (size_t)b * NP * NP + (size_t)i * NP;
  float p[4];
  float sum = 0.0f;
#pragma unroll
  for (int q = 0; q < 4; ++q) {
    p[q] = __expf(lrow[t + q * 256]);
    sum += p[q];
  }
  red[0][t] = sum;
  __syncthreads();
  for (int off = 128; off > 0; off >>= 1) {
    if (t < off) red[0][t] += red[0][t + off];
    __syncthreads();
  }
  const float inv = 1.0f / (red[0][0] + 1e-8f);
  __syncthreads();

  float* prow = perm_out + (size_t)b * N_ * N_ + (size_t)i * N_;
  const float* arow = aff0 + (size_t)b * N_ * N_ + (size_t)i * N_;
  const float* tb = tgt + (size_t)b * 3 * N_;
  float dacc = 0.0f, a0 = 0.0f, a1 = 0.0f, a2 = 0.0f;
#pragma unroll
  for (int q = 0; q < 4; ++q) {
    const int j = t + q * 256;
    const float pn = p[q] * inv;
    prow[j] = pn;
    dacc += pn * arow[j];
    a0 += pn * tb[j];
    a1 += pn * tb[N_ + j];
    a2 += pn * tb[2 * N_ + j];
  }
  red[0][t] = dacc; red[1][t] = a0; red[2][t] = a1; red[3][t] = a2;
  __syncthreads();
  for (int off = 128; off > 0; off >>= 1) {
    if (t < off) {
#pragma unroll
      for (int r = 0; r < 4; ++r) red[r][t] += red[r][t + off];
    }
    __syncthreads();
  }
  if (t == 0) {
    diag[b * N_ + i] = temperature[b] * red[0][0];   // tau * <P, affinity0>
    src_corr[(size_t)b * 3 * N_ + i]            = red[1][0];
    src_corr[(size_t)b * 3 * N_ + N_ + i]       = red[2][0];
    src_corr[(size_t)b * 3 * N_ + 2 * N_ + i]   = red[3][0];
  }
}

// -------------------- rank-1 conv + BN + ReLU MLP (single block) -----------
__global__ __launch_bounds__(256)
void mlp_kernel(const float* __restrict__ diag,   // [B*N] = 8192
                const float* __restrict__ w1, const float* __restrict__ w2,
                const float* __restrict__ g1, const float* __restrict__ b1,
                const float* __restrict__ g2, const float* __restrict__ b2,
                float* __restrict__ xout)         // [B*N]
{
  __shared__ float red[256];
  __shared__ float sw1[64], sw2[64], sg1[64], sb1[64];
  __shared__ float stats[4];
  const int t = threadIdx.x;
  if (t < 64) { sw1[t] = w1[t]; sw2[t] = w2[t]; sg1[t] = g1[t]; sb1[t] = b1[t]; }
  const float invM = 1.0f / (float)(B_ * N_);

  float vals[32];
  float s1 = 0.0f, s2 = 0.0f;
#pragma unroll
  for (int q = 0; q < 32; ++q) {
    const float v = diag[t + q * 256];
    vals[q] = v; s1 += v; s2 += v * v;
  }
  red[t] = s1; __syncthreads();
  for (int off = 128; off > 0; off >>= 1) { if (t < off) red[t] += red[t + off]; __syncthreads(); }
  if (t == 0) stats[0] = red[0] * invM;
  __syncthreads();
  red[t] = s2; __syncthreads();
  for (int off = 128; off > 0; off >>= 1) { if (t < off) red[t] += red[t + off]; __syncthreads(); }
  if (t == 0) stats[1] = fmaxf(red[0] * invM - stats[0] * stats[0], 0.0f);
  __syncthreads();

  const float mu = stats[0], var = stats[1];
  s1 = 0.0f; s2 = 0.0f;
#pragma unroll
  for (int q = 0; q < 32; ++q) {
    const float d = vals[q] - mu;
    float acc = 0.0f;
    for (int c = 0; c < 64; ++c) {
      const float w = sw1[c];
      const float xn = (w * d) * rsqrtf(w * w * var + BN_EPS);
      const float h = fmaxf(xn * sg1[c] + sb1[c], 0.0f);
      acc += sw2[c] * h;
    }
    vals[q] = acc; s1 += acc; s2 += acc * acc;
  }
  red[t] = s1; __syncthreads();
  for (int off = 128; off > 0; off >>= 1) { if (t < off) red[t] += red[t + off]; __syncthreads(); }
  if (t == 0) stats[2] = red[0] * invM;
  __syncthreads();
  red[t] = s2; __syncthreads();
  for (int off = 128; off > 0; off >>= 1) { if (t < off) red[t] += red[t + off]; __syncthreads(); }
  if (t == 0) stats[3] = fmaxf(red[0] * invM - stats[2] * stats[2], 0.0f);
  __syncthreads();

  const float mu2 = stats[2];
  const float irs = rsqrtf(stats[3] + BN_EPS);
  const float G2 = g2[0], Be2 = b2[0];
#pragma unroll
  for (int q = 0; q < 32; ++q)
    xout[t + q * 256] = fmaxf((vals[q] - mu2) * irs * G2 + Be2, 0.0f);
}

// ------ Gumbel straight-through: argmax over n, gather src / src_corr ------
__global__ __launch_bounds__(256)
void gumbel_kernel(const float* __restrict__ x,        // [B][N]
                   const float* __restrict__ gu,       // [B][K][N]
                   const float* __restrict__ src,      // [B][3][N]
                   const float* __restrict__ src_corr, // [B][3][N]
                   float* __restrict__ src_k,          // [B][3][K]
                   float* __restrict__ src_corr_k)     // [B][3][K]
{
  __shared__ float bv[256];
  __shared__ int   bi[256];
  const int b = blockIdx.y, k = blockIdx.x, t = threadIdx.x;
  const float* xb = x + b * N_;
  const float* ub = gu + ((size_t)b * K_ + k) * N_;
  float best = -3.4e38f; int bidx = 0;
#pragma unroll
  for (int q = 0; q < 4; ++q) {
    const int n = t + q * 256;
    const float v = xb[n] - __logf(-__logf(ub[n]));  // x + gumbel
    if (v > best) { best = v; bidx = n; }
  }
  bv[t] = best; bi[t] = bidx;
  __syncthreads();
  for (int off = 128; off > 0; off >>= 1) {
    if (t < off) {
      const float v2 = bv[t + off]; const int i2 = bi[t + off];
      if (v2 > bv[t] || (v2 == bv[t] && i2 < bi[t])) { bv[t] = v2; bi[t] = i2; }
    }
    __syncthreads();
  }
  if (t == 0) {
    const int n = bi[0];
    const float* sb = src + (size_t)b * 3 * N_;
    const float* cb = src_corr + (size_t)b * 3 * N_;
    float* sk = src_k + (size_t)b * 3 * K_;
    float* ck = src_corr_k + (size_t)b * 3 * K_;
    sk[k] = sb[n]; sk[K_ + k] = sb[N_ + n]; sk[2 * K_ + k] = sb[2 * N_ + n];
    ck[k] = cb[n]; ck[K_ + k] = cb[N_ + n]; ck[2 * K_ + k] = cb[2 * N_ + n];
  }
}

// -------------------- covariance + 3x3 Kabsch SVD per batch ----------------
__device__ inline float det3(const float M[3][3]) {
  return M[0][0] * (M[1][1] * M[2][2] - M[1][2] * M[2][1])
       - M[0][1] * (M[1][0] * M[2][2] - M[1][2] * M[2][0])
       + M[0][2] * (M[1][0] * M[2][1] - M[1][1] * M[2][0]);
}

__global__ __launch_bounds__(256)
void kabsch_kernel(const float* __restrict__ src_k,       // [B][3][K]
                   const float* __restrict__ src_corr_k,  // [B][3][K]
                   float* __restrict__ Rout,              // [B][3][3]
                   float* __restrict__ tout)              // [B][3]
{
  __shared__ float red[9][256];
  __shared__ float mu[6];
  const int b = blockIdx.x, t = threadIdx.x;   // one thread per k (K_=256)
  const float* sk = src_k + (size_t)b * 3 * K_;
  const float* ck = src_corr_k + (size_t)b * 3 * K_;
  const float a0 = sk[t], a1 = sk[K_ + t], a2 = sk[2 * K_ + t];
  const float c0 = ck[t], c1 = ck[K_ + t], c2 = ck[2 * K_ + t];
  red[0][t] = a0; red[1][t] = a1; red[2][t] = a2;
  red[3][t] = c0; red[4][t] = c1; red[5][t] = c2;
  __syncthreads();
  for (int off = 128; off > 0; off >>= 1) {
    if (t < off) {
#pragma unroll
      for (int r = 0; r < 6; ++r) red[r][t] += red[r][t + off];
    }
    __syncthreads();
  }
  if (t < 6) mu[t] = red[t][0] * (1.0f / K_);
  __syncthreads();
  const float da0 = a0 - mu[0], da1 = a1 - mu[1], da2 = a2 - mu[2];
  const float db0 = c0 - mu[3], db1 = c1 - mu[4], db2 = c2 - mu[5];
  red[0][t] = da0 * db0; red[1][t] = da0 * db1; red[2][t] = da0 * db2;
  red[3][t] = da1 * db0; red[4][t] = da1 * db1; red[5][t] = da1 * db2;
  red[6][t] = da2 * db0; red[7][t] = da2 * db1; red[8][t] = da2 * db2;
  __syncthreads();
  for (int off = 128; off > 0; off >>= 1) {
    if (t < off) {
#pragma unroll
      for (int r = 0; r < 9; ++r) red[r][t] += red[r][t + off];
    }
    __syncthreads();
  }
  if (t == 0) {
    float H[3][3];
    H[0][0] = red[0][0]; H[0][1] = red[1][0]; H[0][2] = red[2][0];
    H[1][0] = red[3][0]; H[1][1] = red[4][0]; H[1][2] = red[5][0];
    H[2][0] = red[6][0]; H[2][1] = red[7][0]; H[2][2] = red[8][0];
    // S = H^T H, Jacobi eigendecomposition -> V
    float S[3][3];
    for (int i = 0; i < 3; ++i)
      for (int j = 0; j < 3; ++j) {
        float acc = 0.0f;
        for (int r = 0; r < 3; ++r) acc += H[r][i] * H[r][j];
        S[i][j] = acc;
      }
    float V[3][3] = {{1, 0, 0}, {0, 1, 0}, {0, 0, 1}};
    for (int sweep = 0; sweep < 12; ++sweep) {
      for (int pi = 0; pi < 3; ++pi) {
        const int p = (pi == 2) ? 1 : 0;
        const int q = (pi == 0) ? 1 : 2;
        const float apq = S[p][q];
        if (fabsf(apq) < 1e-30f) continue;
        const float theta = (S[q][q] - S[p][p]) / (2.0f * apq);
        const float tt = copysignf(1.0f, theta) /
                         (fabsf(theta) + sqrtf(1.0f + theta * theta));
        const float cc = rsqrtf(1.0f + tt * tt);
        const float sn = tt * cc;
        for (int k = 0; k < 3; ++k) { const float u = S[k][p], v = S[k][q]; S[k][p] = cc * u - sn * v; S[k][q] = sn * u + cc * v; }
        for (int k = 0; k < 3; ++k) { const float u = S[p][k], v = S[q][k]; S[p][k] = cc * u - sn * v; S[q][k] = sn * u + cc * v; }
        for (int k = 0; k < 3; ++k) { const float u = V[k][p], v = V[k][q]; V[k][p] = cc * u - sn * v; V[k][q] = sn * u + cc * v; }
      }
    }
    float lam[3] = {S[0][0], S[1][1], S[2][2]};
    for (int i = 0; i < 2; ++i)
      for (int j = 0; j < 2 - i; ++j)
        if (lam[j] < lam[j + 1]) {
          const float tmp = lam[j]; lam[j] = lam[j + 1]; lam[j + 1] = tmp;
          for (int k = 0; k < 3; ++k) { const float tv = V[k][j]; V[k][j] = V[k][j + 1]; V[k][j + 1] = tv; }
        }
    float U[3][3], sig[3];
    for (int i = 0; i < 3; ++i) {
      const float u0 = H[0][0] * V[0][i] + H[0][1] * V[1][i] + H[0][2] * V[2][i];
      const float u1 = H[1][0] * V[0][i] + H[1][1] * V[1][i] + H[1][2] * V[2][i];
      const float u2 = H[2][0] * V[0][i] + H[2][1] * V[1][i] + H[2][2] * V[2][i];
      const float nn = sqrtf(u0 * u0 + u1 * u1 + u2 * u2);
      sig[i] = nn;
      const float inv = 1.0f / fmaxf(nn, 1e-20f);
      U[0][i] = u0 * inv; U[1][i] = u1 * inv; U[2][i] = u2 * inv;
    }
    if (sig[2] < 1e-6f * fmaxf(sig[0], 1e-30f)) {  // rank-deficient guard
      const float x = U[1][0] * U[2][1] - U[2][0] * U[1][1];
      const float y = U[2][0] * U[0][1] - U[0][0] * U[2][1];
      const float z = U[0][0] * U[1][1] - U[1][0] * U[0][1];
      const float inv = rsqrtf(fmaxf(x * x + y * y + z * z, 1e-30f));
      U[0][2] = x * inv; U[1][2] = y * inv; U[2][2] = z * inv;
    }
    const float d = det3(V) * det3(U);       // det(V U^T)
    const float dv[3] = {1.0f, 1.0f, d};
    float R[3][3];
    for (int i = 0; i < 3; ++i)
      for (int k = 0; k < 3; ++k) {
        float acc = 0.0f;
        for (int j = 0; j < 3; ++j) acc += dv[j] * V[i][j] * U[k][j];
        R[i][k] = acc;
      }
    for (int i = 0; i < 3; ++i) {
      for (int k = 0; k < 3; ++k) Rout[b * 9 + i * 3 + k] = R[i][k];
      tout[b * 3 + i] =
          -(R[i][0] * mu[0] + R[i][1] * mu[1] + R[i][2] * mu[2]) + mu[3 + i];
    }
  }
}

// ---------------------------------------------------------------------------
extern "C" void kernel_launch(void* const* d_in, const int* in_sizes, int n_in,
                              void* d_out, int out_size, void* d_ws, size_t ws_size,
                              hipStream_t stream) {
  (void)in_sizes; (void)n_in; (void)out_size; (void)ws_size;
  const float* srcE = (const float*)d_in[0];
  const float* tgtE = (const float*)d_in[1];
  const float* src  = (const float*)d_in[2];
  const float* tgt  = (const float*)d_in[3];
  const float* temperature = (const float*)d_in[4];
  const float* gumbel_u    = (const float*)d_in[5];
  const float* w1 = (const float*)d_in[6];
  const float* w2 = (const float*)d_in[7];
  const float* g1 = (const float*)d_in[8];
  const float* b1 = (const float*)d_in[9];
  const float* g2 = (const float*)d_in[10];
  const float* b2 = (const float*)d_in[11];

  float* out  = (float*)d_out;
  float* Rout = out;                 // 8*9
  float* tout = out + 72;            // 8*3
  float* Pout = out + 96;            // 8*1024*1024

  float* ws       = (float*)d_ws;
  float* spad     = ws;                                   // B*NP*NP
  float* aff0     = spad + (size_t)B_ * NP * NP;          // B*N*N
  float* diag     = aff0 + (size_t)B_ * N_ * N_;          // B*N
  float* src_corr = diag + (size_t)B_ * N_;               // B*3*N
  float* xmlp     = src_corr + (size_t)B_ * 3 * N_;       // B*N
  float* srck     = xmlp + (size_t)B_ * N_;               // B*3*K
  float* corrk    = srck + (size_t)B_ * 3 * K_;           // B*3*K

  pad_zero_kernel<<<dim3(5, B_), 256, 0, stream>>>(spad);
  affinity_gemm_kernel<<<dim3(N_ / 128, N_ / 64, B_), 256, 0, stream>>>(
      srcE, tgtE, temperature, aff0, spad);
  for (int it = 0; it < 5; ++it) {
    sinkhorn_row_kernel<<<dim3(N_, B_), 256, 0, stream>>>(spad);
    sinkhorn_col_kernel<<<dim3(N_ / 32, B_), 256, 0, stream>>>(spad);
  }
  permnorm_kernel<<<dim3(N_, B_), 256, 0, stream>>>(
      spad, aff0, tgt, temperature, Pout, diag, src_corr);
  mlp_kernel<<<1, 256, 0, stream>>>(diag, w1, w2, g1, b1, g2, b2, xmlp);
  gumbel_kernel<<<dim3(K_, B_), 256, 0, stream>>>(
      xmlp, gumbel_u, src, src_corr, srck, corrk);
  kabsch_kernel<<<B_, 256, 0, stream>>>(srck, corrk, Rout, tout);
}